// Layer_74285754351947
// MI455X (gfx1250) — compile-verified
//
#include <hip/hip_runtime.h>

typedef __attribute__((ext_vector_type(2))) float v2f;
typedef __attribute__((ext_vector_type(8))) float v8f;

#define B_   16
#define C_   256
#define E_   8
#define HW_  4096

#define BM 128
#define BN 128
#define BK 32
#define SA (BK + 4)   // As row stride: 36 floats -> 16B-aligned rows, conflict-free b64 frag loads
#define SB (BK + 2)   // Bs row stride: 34 floats -> 8B-aligned rows, conflict-free b64 frag loads

// ---------------------------------------------------------------------------
// Kernel 1: global average pool  -> pooled[b*C + c]
// ---------------------------------------------------------------------------
__global__ __launch_bounds__(256) void pool_kernel(const float* __restrict__ in,
                                                   float* __restrict__ pooled) {
    __shared__ float red[256];
    const int bc = blockIdx.x;                 // b*C + c
    const float* p = in + (size_t)bc * HW_;
    float s = 0.f;
    for (int i = threadIdx.x; i < HW_; i += 256) s += p[i];
    red[threadIdx.x] = s;
    __syncthreads();
    for (int off = 128; off > 0; off >>= 1) {
        if (threadIdx.x < off) red[threadIdx.x] += red[threadIdx.x + off];
        __syncthreads();
    }
    if (threadIdx.x == 0) pooled[bc] = red[0] * (1.0f / (float)HW_);
}

// ---------------------------------------------------------------------------
// Kernel 2: router logits -> softmax -> top-2 sparse gates (tiny; 1 wave)
// ---------------------------------------------------------------------------
__global__ void gate_kernel(const float* __restrict__ pooled,
                            const float* __restrict__ Wr,
                            const float* __restrict__ br,
                            float* __restrict__ gates) {
    const int b = threadIdx.x;
    if (b >= B_) return;
    float logit[E_];
    for (int e = 0; e < E_; ++e) {
        float s = br[e];
        for (int c = 0; c < C_; ++c) s += pooled[b * C_ + c] * Wr[c * E_ + e];
        logit[e] = s;
    }
    float mx = logit[0];
    for (int e = 1; e < E_; ++e) mx = fmaxf(mx, logit[e]);
    float w[E_], den = 0.f;
    for (int e = 0; e < E_; ++e) { w[e] = __expf(logit[e] - mx); den += w[e]; }
    const float inv = 1.f / den;
    for (int e = 0; e < E_; ++e) w[e] *= inv;
    // top-2 (first index wins ties, matching lax.top_k)
    int i0 = 0;
    for (int e = 1; e < E_; ++e) if (w[e] > w[i0]) i0 = e;
    int i1 = -1;
    for (int e = 0; e < E_; ++e) {
        if (e == i0) continue;
        if (i1 < 0 || w[e] > w[i1]) i1 = e;
    }
    for (int e = 0; e < E_; ++e)
        gates[b * E_ + e] = (e == i0 || e == i1) ? w[e] : 0.f;
}

// ---------------------------------------------------------------------------
// Kernel 3: per-batch combined expert matrix  W_comb[b] = sum_e g_e * W_e
// ---------------------------------------------------------------------------
__global__ __launch_bounds__(256) void combine_kernel(const float* __restrict__ gates,
                                                      const float* __restrict__ Wexp,
                                                      const float* __restrict__ bexp,
                                                      float* __restrict__ wcomb,
                                                      float* __restrict__ bcomb) {
    const int b = blockIdx.x;
    __shared__ float g[E_];
    if (threadIdx.x < E_) g[threadIdx.x] = gates[b * E_ + threadIdx.x];
    __syncthreads();
    float* wout = wcomb + (size_t)b * C_ * C_;
    for (int idx = threadIdx.x; idx < C_ * C_; idx += 256) {
        float s = 0.f;
#pragma unroll
        for (int e = 0; e < E_; ++e) {
            const float ge = g[e];            // uniform branch: only 2 experts live
            if (ge != 0.f) s += ge * Wexp[(size_t)e * C_ * C_ + idx];
        }
        wout[idx] = s;
    }
    const int d = threadIdx.x;
    float sb = 0.f;
#pragma unroll
    for (int e = 0; e < E_; ++e) sb += g[e] * bexp[e * C_ + d];
    bcomb[b * C_ + d] = sb;
}

// ---------------------------------------------------------------------------
// Kernel 4: batched GEMM  out[b] = x[b] + W_comb[b] @ x[b] + bias[b]
// M=256, K=256, N=4096 per batch.  128x128 tiles, BK=32, 8 waves/block.
// idx-major LDS layouts so every WMMA fragment is ONE contiguous ds_load_b64
// into an even-aligned VGPR pair (no v_mov shuffles before v_wmma).
// ---------------------------------------------------------------------------
__global__ __launch_bounds__(256) void moe_gemm(const float* __restrict__ in,
                                                const float* __restrict__ wcomb,
                                                const float* __restrict__ bcomb,
                                                float* __restrict__ out) {
    __shared__ __align__(16) float As[BM][SA];   // As[m][k]
    __shared__ __align__(16) float Bs[BN][SB];   // Bs[n][k]

    const int b      = blockIdx.z;
    const int tile_m = blockIdx.y * BM;
    const int tile_n = blockIdx.x * BN;

    const float* A  = wcomb + (size_t)b * C_ * C_;   // [256 x 256] row-major (d,c)
    const float* Bg = in    + (size_t)b * C_ * HW_;  // [256 x 4096] row-major (c,hw)

    const int tid  = threadIdx.x;
    const int wave = tid >> 5;
    const int lane = tid & 31;
    const int lr   = lane & 15;    // lane-in-halfwave (stripes M / N)
    const int lh   = lane >> 4;    // half select (splits K: {0,1} vs {2,3})
    const int wrow = wave & 3;     // 4 wave-rows  -> 32 rows each
    const int wcol = wave >> 2;    // 2 wave-cols  -> 64 cols each

    v8f acc[2][4] = {};            // zero accumulators

    for (int k0 = 0; k0 < C_; k0 += BK) {
        // ---- stage A tile (BM x BK): direct float4 copy, k-contiguous ----
#pragma unroll
        for (int i = 0; i < 4; ++i) {
            const int idx = tid + i * 256;
            const int row = idx >> 3;          // 8 float4 per 32-float row
            const int c4  = (idx & 7) << 2;
            *reinterpret_cast<float4*>(&As[row][c4]) =
                *reinterpret_cast<const float4*>(
                    A + (size_t)(tile_m + row) * C_ + k0 + c4);
        }
        // ---- stage B tile (BK x BN): transposed scatter into Bs[n][k] ----
#pragma unroll
        for (int i = 0; i < 4; ++i) {
            const int idx  = tid + i * 256;
            const int krow = idx >> 5;         // 32 float4 per 128-float row
            const int n4   = (idx & 31) << 2;
            const float4 v = *reinterpret_cast<const float4*>(
                Bg + (size_t)(k0 + krow) * HW_ + tile_n + n4);
            Bs[n4 + 0][krow] = v.x;
            Bs[n4 + 1][krow] = v.y;
            Bs[n4 + 2][krow] = v.z;
            Bs[n4 + 3][krow] = v.w;
        }
        // ---- prefetch next K-tile while this one is consumed ----
        if (k0 + BK < C_) {
            if (tid < 128) {                   // next A tile: one 128B line per row
                __builtin_prefetch(A + (size_t)(tile_m + tid) * C_ + k0 + BK, 0, 1);
            } else {                           // next B tile: 4 lines per k-row
                const int t   = tid - 128;
                const int row = t >> 2;
                const int seg = t & 3;
                __builtin_prefetch(Bg + (size_t)(k0 + BK + row) * HW_ + tile_n + seg * 32, 0, 1);
            }
        }
        __syncthreads();

#pragma unroll
        for (int kk = 0; kk < BK; kk += 4) {
            const int kfrag = kk + lh * 2;     // this half-wave's K pair
            v2f af[2];
#pragma unroll
            for (int mi = 0; mi < 2; ++mi) {
                const int m = wrow * 32 + mi * 16 + lr;
                af[mi] = *reinterpret_cast<const v2f*>(&As[m][kfrag]);
            }
            v2f bf[4];
#pragma unroll
            for (int ni = 0; ni < 4; ++ni) {
                const int n = wcol * 64 + ni * 16 + lr;
                bf[ni] = *reinterpret_cast<const v2f*>(&Bs[n][kfrag]);
            }
#pragma unroll
            for (int mi = 0; mi < 2; ++mi)
#pragma unroll
                for (int ni = 0; ni < 4; ++ni)
                    acc[mi][ni] = __builtin_amdgcn_wmma_f32_16x16x4_f32(
                        false, af[mi], false, bf[ni],
                        (short)0, acc[mi][ni], false, false);
        }
        __syncthreads();
    }

    // ---- epilogue: residual + bias, C/D layout VGPR j -> M = j + lh*8 ----
    const float* inb  = in  + (size_t)b * C_ * HW_;
    float*       outb = out + (size_t)b * C_ * HW_;
#pragma unroll
    for (int mi = 0; mi < 2; ++mi) {
#pragma unroll
        for (int j = 0; j < 8; ++j) {
            const int d    = tile_m + wrow * 32 + mi * 16 + lh * 8 + j;
            const float bv = bcomb[b * C_ + d];
#pragma unroll
            for (int ni = 0; ni < 4; ++ni) {
                const int hw = tile_n + wcol * 64 + ni * 16 + lr;
                outb[(size_t)d * HW_ + hw] =
                    inb[(size_t)d * HW_ + hw] + acc[mi][ni][j] + bv;
            }
        }
    }
}

// ---------------------------------------------------------------------------
extern "C" void kernel_launch(void* const* d_in, const int* in_sizes, int n_in,
                              void* d_out, int out_size, void* d_ws, size_t ws_size,
                              hipStream_t stream) {
    const float* in   = (const float*)d_in[0];  // [16,256,64,64]
    const float* Wr   = (const float*)d_in[1];  // [256,8]
    const float* br   = (const float*)d_in[2];  // [8]
    const float* Wexp = (const float*)d_in[3];  // [8,256,256]
    const float* bexp = (const float*)d_in[4];  // [8,256]
    float* out = (float*)d_out;

    // workspace layout (floats)
    float* ws     = (float*)d_ws;
    float* pooled = ws;                         // 16*256      = 4096
    float* gates  = ws + 4096;                  // 16*8        = 128
    float* bcomb  = ws + 4096 + 128;            // 16*256      = 4096
    float* wcomb  = ws + 4096 + 128 + 4096;     // 16*256*256  = 1,048,576 (16B-aligned)

    pool_kernel<<<B_ * C_, 256, 0, stream>>>(in, pooled);
    gate_kernel<<<1, 32, 0, stream>>>(pooled, Wr, br, gates);
    combine_kernel<<<B_, 256, 0, stream>>>(gates, Wexp, bexp, wcomb, bcomb);

    dim3 grid(HW_ / BN, C_ / BM, B_);           // 32 x 2 x 16 = 1024 blocks
    moe_gemm<<<grid, 256, 0, stream>>>(in, wcomb, bcomb, out);
}